// AttentionDistillationLoss_10806137717453
// MI455X (gfx1250) — compile-verified
//
#include <hip/hip_runtime.h>
#include <hip/hip_bf16.h>
#include <math.h>

// Problem constants (from the reference)
#define BI   256
#define LI1  37
#define KD   256
#define BS   256
#define LS1  31
#define LIV  36   // Li = LI1-1
#define LSV  30   // Ls = LS1-1

typedef __attribute__((ext_vector_type(2))) float v2f;
typedef __attribute__((ext_vector_type(8))) float v8f;

#define XSTRIDE 132   // 128 K-chunk + 4 pad -> bank (4*row + k) % 64, conflict-free
#define ALSTR   52

__device__ __forceinline__ float wave_sum32(float v) {
    #pragma unroll
    for (int m = 16; m > 0; m >>= 1) v += __shfl_xor(v, m, 32);
    return v;
}
__device__ __forceinline__ float wave_max32(float v) {
    #pragma unroll
    for (int m = 16; m > 0; m >>= 1) v = fmaxf(v, __shfl_xor(v, m, 32));
    return v;
}

// CDNA5 async global->LDS DMA (B128). LDS offset = low 32 bits of generic ptr,
// global address = 64-bit VGPR pair. Tracked by ASYNCcnt.
__device__ __forceinline__ void async_load_b128(void* lds_ptr, const void* gptr) {
    unsigned int lds_off = (unsigned int)(uintptr_t)lds_ptr;
    unsigned long long ga = (unsigned long long)(uintptr_t)gptr;
    asm volatile("global_load_async_to_lds_b128 %0, %1, off"
                 :: "v"(lds_off), "v"(ga)
                 : "memory");
}
__device__ __forceinline__ void wait_async_zero() {
    asm volatile("s_wait_asynccnt 0x0" ::: "memory");
}

__global__ __launch_bounds__(192)
void adl_pair_kernel(const float* __restrict__ im_set,
                     const float* __restrict__ s_seq,
                     const int*   __restrict__ im_len,
                     const int*   __restrict__ s_len,
                     const float* __restrict__ teacher,
                     float* __restrict__ partials,
                     int mode) {
    __shared__ float xs[48 * XSTRIDE];   // B operand: x[i, p, k-chunk], rows 36..47 zero pad
    __shared__ float ys[32 * XSTRIDE];   // A operand: y[j, q, k-chunk], rows 30..31 zero pad
    __shared__ float al[32 * ALSTR];     // align tile (q x p), fp32
    __shared__ float wsum[6];

    const int i = blockIdx.x;
    const int j = blockIdx.y;
    const int tid  = threadIdx.x;          // 0..191, 6 waves
    const int wv   = tid >> 5;
    const int lane = tid & 31;
    const int half = lane >> 4;
    const int l    = lane & 15;
    const int mt = wv / 3;                 // q-tile 0..1
    const int nt = wv % 3;                 // p-tile 0..2
    const int arow = mt * 16 + l;          // row of Y (q)
    const int bcol = nt * 16 + l;          // row of X (p) == column of B

    const float* __restrict__ tbase =
        teacher + (((size_t)i * BS + j) * (size_t)LSV) * (size_t)LIV;

    // Prefetch the 30*36*4 = 4320-byte teacher block while the WMMA phase runs.
    {
        const int tbytes = LSV * LIV * 4;
        int off = tid * 128;
        if (off < tbytes)
            __builtin_prefetch((const char*)tbase + off, 0, 1);
    }

    // Zero-fill the pad rows ONCE (they are never touched by async loads, and
    // both K-chunks reuse the same valid-row region).
    {
        const float4 z = make_float4(0.f, 0.f, 0.f, 0.f);
        for (int idx = tid; idx < (12 * 32 + 2 * 32); idx += 192) {
            if (idx < 12 * 32) {
                const int row = LIV + (idx >> 5);
                const int c4  = (idx & 31) << 2;
                *(float4*)&xs[row * XSTRIDE + c4] = z;
            } else {
                const int t2  = idx - 12 * 32;
                const int row = LSV + (t2 >> 5);
                const int c4  = (t2 & 31) << 2;
                *(float4*)&ys[row * XSTRIDE + c4] = z;
            }
        }
    }

    v8f acc = {};

    #pragma unroll
    for (int chunk = 0; chunk < 2; ++chunk) {
        const int c0 = chunk * 128;

        // Async-stage X chunk: 36 rows x 32 float4 = 1152 -> exactly 6 per thread
        const float* xsrc = im_set + ((size_t)i * LI1 + 1) * KD + c0;
        #pragma unroll
        for (int it = 0; it < 6; ++it) {
            const int idx = tid + it * 192;
            const int row = idx >> 5;
            const int c4  = (idx & 31) << 2;
            async_load_b128(&xs[row * XSTRIDE + c4], xsrc + (size_t)row * KD + c4);
        }
        // Async-stage Y chunk: 30 rows x 32 float4 = 960 -> exactly 5 per thread
        const float* ysrc = s_seq + ((size_t)j * LS1 + 1) * KD + c0;
        #pragma unroll
        for (int it = 0; it < 5; ++it) {
            const int idx = tid + it * 192;
            const int row = idx >> 5;
            const int c4  = (idx & 31) << 2;
            async_load_b128(&ys[row * XSTRIDE + c4], ysrc + (size_t)row * KD + c4);
        }
        wait_async_zero();
        __syncthreads();

        // 32 fp32 WMMA steps over this K chunk: D += A(16x4) * B(4x16)
        #pragma unroll 4
        for (int kb = 0; kb < 128; kb += 4) {
            const int k0 = kb + half * 2;  // lanes 16-31 carry K+2,K+3 (ISA f32 A/B layout)
            v2f a = *(const v2f*)&ys[arow * XSTRIDE + k0];
            v2f b = *(const v2f*)&xs[bcol * XSTRIDE + k0];
            acc = __builtin_amdgcn_wmma_f32_16x16x4_f32(
                false, a, false, b, (short)0, acc, false, false);
        }
        __syncthreads();
    }

    // Scatter accumulator to LDS align buffer (scale by 1/sqrt(K) = 1/16)
    #pragma unroll
    for (int r = 0; r < 8; ++r) {
        const int m = mt * 16 + r + half * 8;   // C/D layout: VGPR r -> M=r (+8 for hi half)
        const int n = nt * 16 + l;
        al[m * ALSTR + n] = acc[r] * 0.0625f;
    }
    __syncthreads();

    // ---- Phase 2: masked log-softmax over p + teacher-normalized KL ----
    const float NEGINF = -__builtin_inff();
    const int il  = im_len[i] - 1;   // valid p count
    const int slj = s_len[j] - 1;    // valid q count

    float wacc = 0.f;
    for (int q = wv; q < LSV; q += 6) {
        const int p1 = lane;
        const int p2 = lane + 32;
        const bool in2 = (p2 < LIV);
        float v1 = (p1 < il)         ? al[q * ALSTR + p1] : NEGINF;
        float v2 = (in2 && p2 < il)  ? al[q * ALSTR + p2] : NEGINF;

        const float m = wave_max32(fmaxf(v1, v2));
        const float e = expf(v1 - m) + (in2 ? expf(v2 - m) : 0.f);
        const float Z = wave_sum32(e);
        const float logZ = m + logf(Z);

        const float* tr = tbase + (size_t)q * LIV;
        const float t1 = tr[p1];                 // p1 < 32 < 36 always valid
        const float t2 = in2 ? tr[p2] : 0.f;
        const float tsum = wave_sum32(fabsf(t1) + fabsf(t2));
        const float denom = fmaxf(tsum, 1e-12f);
        const float tn1 = t1 / denom;
        const float tn2 = t2 / denom;

        float kl = 0.f;
        if (tn1 > 0.f)        kl += tn1 * (logf(tn1) - (v1 - logZ));
        if (in2 && tn2 > 0.f) kl += tn2 * (logf(tn2) - (v2 - logZ));
        const float rkl = wave_sum32(kl);

        if (lane == 0 && q < slj) wacc += rkl;
    }

    if (lane == 0) wsum[wv] = wacc;
    __syncthreads();
    if (tid == 0) {
        float total = 0.f;
        #pragma unroll
        for (int w = 0; w < 6; ++w) total += wsum[w];
        if (mode == 0) partials[(size_t)i * BS + j] = total;
        else           atomicAdd(partials, total);
    }
}

__global__ void adl_zero_kernel(float* p) {
    if (threadIdx.x == 0) p[0] = 0.f;
}

__global__ void adl_finalize_kernel(const float* __restrict__ partials,
                                    const int*   __restrict__ s_len,
                                    float* __restrict__ out,
                                    int mode) {
    __shared__ float red[256];
    __shared__ int   ired[256];
    const int tid = threadIdx.x;

    float local = 0.f;
    if (mode == 0) {
        for (int idx = tid; idx < BI * BS; idx += 256) local += partials[idx];
    } else if (tid == 0) {
        local = partials[0];
    }
    int c = s_len[tid] - 1;
    c = c < 0 ? 0 : (c > LSV ? LSV : c);

    red[tid]  = local;
    ired[tid] = c;
    __syncthreads();
    for (int s = 128; s > 0; s >>= 1) {
        if (tid < s) { red[tid] += red[tid + s]; ired[tid] += ired[tid + s]; }
        __syncthreads();
    }
    if (tid == 0) {
        const float n_rows = 256.0f * (float)ired[0];
        out[0] = red[0] / n_rows;
    }
}

extern "C" void kernel_launch(void* const* d_in, const int* in_sizes, int n_in,
                              void* d_out, int out_size, void* d_ws, size_t ws_size,
                              hipStream_t stream) {
    const float* im_set  = (const float*)d_in[0];
    const float* s_seq   = (const float*)d_in[1];
    const int*   im_len  = (const int*)d_in[2];
    const int*   s_len   = (const int*)d_in[3];
    const float* teacher = (const float*)d_in[4];
    float* out = (float*)d_out;
    float* ws  = (float*)d_ws;

    const size_t needed = (size_t)BI * BS * sizeof(float);
    const int mode = (ws_size >= needed) ? 0 : 1;   // 0: per-pair partials, 1: atomic fallback

    if (mode == 1) adl_zero_kernel<<<1, 32, 0, stream>>>(ws);

    adl_pair_kernel<<<dim3(BI, BS), 192, 0, stream>>>(
        im_set, s_seq, im_len, s_len, teacher, ws, mode);

    adl_finalize_kernel<<<1, 256, 0, stream>>>(ws, s_len, out, mode);
}